// multi_GCN_34643206210129
// MI455X (gfx1250) — compile-verified
//
#include <hip/hip_runtime.h>
#include <math.h>

#define N_NODES 50000
#define N_EDGES 1600000
#define IN_SIZE 128
#define HIDDEN  64
#define N_CLASS 16
#define N_TILES (N_NODES / 16)   // 3125, exact

typedef __attribute__((ext_vector_type(2))) float v2f;
typedef __attribute__((ext_vector_type(8))) float v8f;

// ---------------------------------------------------------------- utilities

__global__ void zero_ws_kernel(float* __restrict__ w, long n) {
  long i = (long)blockIdx.x * blockDim.x + threadIdx.x;
  long stride = (long)gridDim.x * blockDim.x;
  for (; i < n; i += stride) w[i] = 0.0f;
}

// deg_out over src, deg_in over dst (float adds of 1.0 are exact < 2^24)
__global__ void degree_kernel(const int* __restrict__ src, const int* __restrict__ dst,
                              float* __restrict__ dego, float* __restrict__ degi) {
  int e = blockIdx.x * blockDim.x + threadIdx.x;
  if (e < N_EDGES) {
    atomicAdd(&dego[src[e]], 1.0f);
    atomicAdd(&degi[dst[e]], 1.0f);
  }
}

__global__ void to_norm_kernel(float* __restrict__ d, int n) {
  int i = blockIdx.x * blockDim.x + threadIdx.x;
  if (i < n) d[i] = rsqrtf(fmaxf(d[i], 1.0f));
}

// ------------------------------------------------- xw = (x * norm_out) @ W
// f32 WMMA 16x16x4. One wave per 16-row tile, 4 N-tiles (HIDDEN=64), K=128.
// A frag (16x4): lanes 0-15 hold K=k4+0,k4+1 of row M0+lane; lanes 16-31 hold
// K=k4+2,k4+3. B frag (4x16): same K split, N = lane&15.
__global__ void gemm_xw_kernel(const float* __restrict__ x,
                               const float* __restrict__ W1, const float* __restrict__ W2,
                               const float* __restrict__ no1, const float* __restrict__ no2,
                               float* __restrict__ xw1, float* __restrict__ xw2) {
  const float* W  = blockIdx.y ? W2  : W1;
  const float* no = blockIdx.y ? no2 : no1;
  float* xw       = blockIdx.y ? xw2 : xw1;

  int wave = threadIdx.x >> 5;
  int tile = blockIdx.x * (blockDim.x >> 5) + wave;
  if (tile >= N_TILES) return;                 // wave-uniform: EXEC stays all-1s
  int lane = threadIdx.x & 31;
  int half = lane >> 4, l16 = lane & 15;
  int row0 = tile * 16;
  int arow = row0 + l16;
  float scale = no[arow];
  const float* xr = x + (long)arow * IN_SIZE;

  v8f acc0 = {}, acc1 = {}, acc2 = {}, acc3 = {};
  for (int k4 = 0; k4 < IN_SIZE; k4 += 4) {
    int ka = k4 + 2 * half;
    float2 av = *(const float2*)(xr + ka);
    v2f a; a.x = av.x * scale; a.y = av.y * scale;
    const float* wr0 = W + (long)ka * HIDDEN;
    const float* wr1 = wr0 + HIDDEN;
    v2f b0; b0.x = wr0[l16];      b0.y = wr1[l16];
    v2f b1; b1.x = wr0[16 + l16]; b1.y = wr1[16 + l16];
    v2f b2; b2.x = wr0[32 + l16]; b2.y = wr1[32 + l16];
    v2f b3; b3.x = wr0[48 + l16]; b3.y = wr1[48 + l16];
    acc0 = __builtin_amdgcn_wmma_f32_16x16x4_f32(false, a, false, b0, (short)0, acc0, false, false);
    acc1 = __builtin_amdgcn_wmma_f32_16x16x4_f32(false, a, false, b1, (short)0, acc1, false, false);
    acc2 = __builtin_amdgcn_wmma_f32_16x16x4_f32(false, a, false, b2, (short)0, acc2, false, false);
    acc3 = __builtin_amdgcn_wmma_f32_16x16x4_f32(false, a, false, b3, (short)0, acc3, false, false);
  }
#pragma unroll
  for (int j = 0; j < 8; ++j) {
    int r = row0 + j + 8 * half;               // C/D layout: vgpr j, half-wave row split
    float* o = xw + (long)r * HIDDEN + l16;
    o[0]  = acc0[j];
    o[16] = acc1[j];
    o[32] = acc2[j];
    o[48] = acc3[j];
  }
}

// --------------------------------------- edge scatter: h[dst] += xw[src]
// One wave per edge: 64 f32 row = 2 per lane (b64 load + 2 f32 atomics).
// xw (12.8 MB) and h (25.6 MB) are L2-resident on MI455X (192 MB L2).
__global__ void scatter_kernel(const int* __restrict__ src, const int* __restrict__ dst,
                               const float* __restrict__ xw, float* __restrict__ h,
                               int col_off) {
  int e = blockIdx.x * (blockDim.x >> 5) + (threadIdx.x >> 5);
  if (e >= N_EDGES) return;
  int lane = threadIdx.x & 31;
  int s = src[e], d = dst[e];
  float2 v = *(const float2*)(xw + (long)s * HIDDEN + lane * 2);
  float* hp = h + (long)d * (2 * HIDDEN) + col_off + lane * 2;
  atomicAdd(hp,     v.x);
  atomicAdd(hp + 1, v.y);
}

// ---------------- out = log_softmax( relu(agg*ni + b) @ Wout + bout )
// A-fragment load fuses norm_in, bias and ReLU (saves one full pass over h).
// K pairs (ka, ka+1) never straddle the layer-1/layer-2 column boundary (64).
__global__ void gemm_out_kernel(const float* __restrict__ h,
                                const float* __restrict__ ni1, const float* __restrict__ ni2,
                                const float* __restrict__ b1,  const float* __restrict__ b2,
                                const float* __restrict__ Wout, const float* __restrict__ bout,
                                float* __restrict__ out) {
  int wave = threadIdx.x >> 5;
  int tile = blockIdx.x * (blockDim.x >> 5) + wave;
  if (tile >= N_TILES) return;
  int lane = threadIdx.x & 31;
  int half = lane >> 4, l16 = lane & 15;
  int row0 = tile * 16;
  int arow = row0 + l16;
  float n1 = ni1[arow], n2 = ni2[arow];
  const float* hr = h + (long)arow * (2 * HIDDEN);

  v8f acc = {};
  for (int k4 = 0; k4 < 2 * HIDDEN; k4 += 4) {
    int ka = k4 + 2 * half;
    float sc = (ka < HIDDEN) ? n1 : n2;
    const float* bb = (ka < HIDDEN) ? b1 : b2;
    int c = ka & (HIDDEN - 1);
    float2 hv = *(const float2*)(hr + ka);
    v2f a;
    a.x = fmaxf(hv.x * sc + bb[c],     0.0f);
    a.y = fmaxf(hv.y * sc + bb[c + 1], 0.0f);
    v2f b;
    b.x = Wout[(long)ka * N_CLASS + l16];
    b.y = Wout[(long)(ka + 1) * N_CLASS + l16];
    acc = __builtin_amdgcn_wmma_f32_16x16x4_f32(false, a, false, b, (short)0, acc, false, false);
  }

  float bo = bout[l16];
#pragma unroll
  for (int j = 0; j < 8; ++j) {
    float v = acc[j] + bo;
    // row max / sum over the 16 classes: xor masks 1,2,4,8 stay within each
    // 16-lane half, matching the C-matrix half-wave row split.
    float m = v;
    m = fmaxf(m, __shfl_xor(m, 1));
    m = fmaxf(m, __shfl_xor(m, 2));
    m = fmaxf(m, __shfl_xor(m, 4));
    m = fmaxf(m, __shfl_xor(m, 8));
    float s = expf(v - m);
    s += __shfl_xor(s, 1);
    s += __shfl_xor(s, 2);
    s += __shfl_xor(s, 4);
    s += __shfl_xor(s, 8);
    int r = row0 + j + 8 * half;
    out[(long)r * N_CLASS + l16] = v - m - logf(s);
  }
}

// ---------------------------------------------------------------- driver

extern "C" void kernel_launch(void* const* d_in, const int* in_sizes, int n_in,
                              void* d_out, int out_size, void* d_ws, size_t ws_size,
                              hipStream_t stream) {
  const float* x    = (const float*)d_in[0];
  const int*   src1 = (const int*)  d_in[1];
  const int*   dst1 = (const int*)  d_in[2];
  const int*   src2 = (const int*)  d_in[3];
  const int*   dst2 = (const int*)  d_in[4];
  const float* W1   = (const float*)d_in[5];
  const float* b1   = (const float*)d_in[6];
  const float* W2   = (const float*)d_in[7];
  const float* b2   = (const float*)d_in[8];
  const float* Wout = (const float*)d_in[9];
  const float* bout = (const float*)d_in[10];
  float* out = (float*)d_out;

  // workspace layout (floats): 4 norm arrays, xw1, xw2, concat accumulator h
  // total = 50000*(4 + 64 + 64 + 128) * 4 B = 52 MB
  float* ws  = (float*)d_ws;
  float* no1 = ws;
  float* ni1 = ws + 1L * N_NODES;
  float* no2 = ws + 2L * N_NODES;
  float* ni2 = ws + 3L * N_NODES;
  float* xw1 = ws + 4L * N_NODES;
  float* xw2 = xw1 + (long)N_NODES * HIDDEN;
  float* h   = xw2 + (long)N_NODES * HIDDEN;
  long total_ws = 4L * N_NODES + 2L * N_NODES * HIDDEN + (long)N_NODES * 2 * HIDDEN;

  // 1. zero accumulators (every call: launch must be replay-deterministic)
  zero_ws_kernel<<<2048, 256, 0, stream>>>(ws, total_ws);

  // 2. degrees (atomic float increments)
  degree_kernel<<<N_EDGES / 256, 256, 0, stream>>>(src1, dst1, no1, ni1);
  degree_kernel<<<N_EDGES / 256, 256, 0, stream>>>(src2, dst2, no2, ni2);

  // 3. norms in place
  to_norm_kernel<<<(4 * N_NODES + 255) / 256, 256, 0, stream>>>(ws, 4 * N_NODES);

  // 4. xw = (x * norm_out) @ W, both layers (grid.y)
  dim3 g4((N_TILES + 7) / 8, 2);
  gemm_xw_kernel<<<g4, 256, 0, stream>>>(x, W1, W2, no1, no2, xw1, xw2);

  // 5. edge scatter into concat buffer (h1 -> cols 0..63, h2 -> cols 64..127)
  scatter_kernel<<<N_EDGES / 8, 256, 0, stream>>>(src1, dst1, xw1, h, 0);
  scatter_kernel<<<N_EDGES / 8, 256, 0, stream>>>(src2, dst2, xw2, h, HIDDEN);

  // 6. fused norm_in + bias + relu + GEMM + log_softmax
  gemm_out_kernel<<<(N_TILES + 7) / 8, 256, 0, stream>>>(h, ni1, ni2, b1, b2,
                                                         Wout, bout, out);
}